// ReadUnit_37769942401462
// MI455X (gfx1250) — compile-verified
//
#include <hip/hip_runtime.h>
#include <math.h>

// ReadUnit fused kernels for MI455X (gfx1250, wave32).
//
// Algebra: rai[b,n] = sum_d p[b,d]*kb_proj[b,d,n] + sum_d q[b,d]*kb[b,d,n] (+const)
//   p[b,d] = u[b,d]*mem[b,d],  q[b,d] = u[b,D+d]
//   mem = memory_state @ W_mem^T + b_mem          (fp32 WMMA GEMM)
//   u   = (ctrl * W_attn) @ W_cat                 (fp32 WMMA GEMM)
// Constants over n (b_cat, b_attn terms) vanish under softmax.

typedef __attribute__((ext_vector_type(2))) float v2f;
typedef __attribute__((ext_vector_type(4))) float v4f;
typedef __attribute__((ext_vector_type(8))) float v8f;

#define B_ 512
#define D_ 512
#define N_ 196

// ---------------------------------------------------------------------------
// Kernel 1: mem[b,d] = sum_k MS[b,k]*W_mem[d,k] + b_mem[d]
// One wave per 16x16 output tile, V_WMMA_F32_16X16X4_F32, K-loop of 128.
// Tiles: (512/16)*(512/16) = 1024 -> 256 blocks x 4 waves.
// ---------------------------------------------------------------------------
__global__ __launch_bounds__(128)
void mem_proj_kernel(const float* __restrict__ MS, const float* __restrict__ Wm,
                     const float* __restrict__ bm, float* __restrict__ mem)
{
    const int wave = threadIdx.x >> 5;
    const int lane = threadIdx.x & 31;
    const int tile = blockIdx.x * 4 + wave;
    const int tM = tile >> 5;          // 32 tiles along N(=D)
    const int tN = tile & 31;
    const int h = lane >> 4;           // half-wave select
    const int m = lane & 15;

    // A-frag layout: lane(h,m) holds A[m][k+2h], A[m][k+2h+1]
    // B-frag layout: lane(h,m) holds B[k+2h][m], B[k+2h+1][m]; B[k][n]=W_mem[n][k]
    const float* pA = MS + (tM * 16 + m) * D_ + 2 * h;
    const float* pB = Wm + (tN * 16 + m) * D_ + 2 * h;

    v8f c = {};
    #pragma unroll 8
    for (int k = 0; k < D_; k += 4) {
        v2f a = *(const v2f*)(pA + k);
        v2f b = *(const v2f*)(pB + k);
        c = __builtin_amdgcn_wmma_f32_16x16x4_f32(false, a, false, b,
                                                  (short)0, c, false, false);
    }
    // C/D layout: c[r] = C[tM*16 + r + 8h][tN*16 + m]
    const float bias = bm[tN * 16 + m];
    float* pC = mem + (tM * 16 + 8 * h) * D_ + tN * 16 + m;
    #pragma unroll
    for (int r = 0; r < 8; ++r)
        pC[r * D_] = c[r] + bias;
}

// ---------------------------------------------------------------------------
// Kernel 2: u[b,j] = sum_d (ctrl[b,d]*W_attn[d]) * W_cat[d,j],  j in [0,2D)
// A fused on the fly; W_attn staged in LDS. Tiles: 32*64 = 2048 -> 512 blocks.
// ---------------------------------------------------------------------------
__global__ __launch_bounds__(128)
void u_proj_kernel(const float* __restrict__ ctrl, const float* __restrict__ Wa,
                   const float* __restrict__ Wcat, float* __restrict__ u)
{
    __shared__ __align__(16) float wa[D_];
    for (int i = threadIdx.x; i < D_; i += 128) wa[i] = Wa[i];
    __syncthreads();

    const int wave = threadIdx.x >> 5;
    const int lane = threadIdx.x & 31;
    const int tile = blockIdx.x * 4 + wave;
    const int tM = tile >> 6;          // 64 tiles along N(=2D)
    const int tN = tile & 63;
    const int h = lane >> 4;
    const int m = lane & 15;

    const int col = tN * 16 + m;                       // output column j
    const float* pA = ctrl + (tM * 16 + m) * D_ + 2 * h;
    const float* pB = Wcat + 2 * h * (2 * D_) + col;   // row k+2h of W_cat

    v8f c = {};
    #pragma unroll 4
    for (int k = 0; k < D_; k += 4) {
        const int kk = k + 2 * h;
        v2f ct = *(const v2f*)(pA + k);
        v2f a;
        a.x = ct.x * wa[kk];
        a.y = ct.y * wa[kk + 1];
        v2f b;
        b.x = pB[k * (2 * D_)];                        // W_cat[kk][col]
        b.y = pB[(k + 1) * (2 * D_)];                  // W_cat[kk+1][col]
        c = __builtin_amdgcn_wmma_f32_16x16x4_f32(false, a, false, b,
                                                  (short)0, c, false, false);
    }
    float* pC = u + (tM * 16 + 8 * h) * (2 * D_) + col;
    #pragma unroll
    for (int r = 0; r < 8; ++r)
        pC[r * (2 * D_)] = c[r];
}

// ---------------------------------------------------------------------------
// Kernel 3 (bandwidth bound): per batch b,
//   rai[n]  = sum_d p[d]*kb_proj[b,d,n] + q[d]*kb[b,d,n]
//   rvi     = softmax_n(rai)
//   out[b,d]= sum_n rvi[n]*kb[b,d,n]
// One block (256 thr / 8 waves) per batch; p,q,rvi broadcast from LDS.
// ---------------------------------------------------------------------------
__global__ __launch_bounds__(256)
void attend_kernel(const float* __restrict__ kb, const float* __restrict__ kbp,
                   const float* __restrict__ memb, const float* __restrict__ ub,
                   float* __restrict__ out)
{
    __shared__ __align__(16) float p[D_];
    __shared__ __align__(16) float q[D_];
    __shared__ __align__(16) float red[256];
    __shared__ __align__(16) float w[256];

    const int b = blockIdx.x;
    const int t = threadIdx.x;

    // prologue: p = u[:D]*mem, q = u[D:2D]
    for (int d = t; d < D_; d += 256) {
        p[d] = ub[b * 2 * D_ + d] * memb[b * D_ + d];
        q[d] = ub[b * 2 * D_ + D_ + d];
    }
    __syncthreads();

    // ---- pass 1: logits; lane owns one n-column (coalesced 128B row reads)
    float acc = 0.0f;
    if (t < N_) {
        const float* kpb = kbp + (size_t)b * D_ * N_ + t;
        const float* kbb = kb  + (size_t)b * D_ * N_ + t;
        #pragma unroll 4
        for (int d = 0; d < D_; d += 4) {
            v4f pv = *(const v4f*)(p + d);
            v4f qv = *(const v4f*)(q + d);
            acc += pv.x * kpb[(d    ) * N_] + qv.x * kbb[(d    ) * N_];
            acc += pv.y * kpb[(d + 1) * N_] + qv.y * kbb[(d + 1) * N_];
            acc += pv.z * kpb[(d + 2) * N_] + qv.z * kbb[(d + 2) * N_];
            acc += pv.w * kpb[(d + 3) * N_] + qv.w * kbb[(d + 3) * N_];
        }
    }

    // ---- softmax over n (block tree reduction)
    red[t] = (t < N_) ? acc : -INFINITY;
    __syncthreads();
    #pragma unroll
    for (int s = 128; s > 0; s >>= 1) {
        if (t < s) red[t] = fmaxf(red[t], red[t + s]);
        __syncthreads();
    }
    const float mx = red[0];
    __syncthreads();
    const float e = (t < N_) ? __expf(acc - mx) : 0.0f;
    red[t] = e;
    __syncthreads();
    #pragma unroll
    for (int s = 128; s > 0; s >>= 1) {
        if (t < s) red[t] += red[t + s];
        __syncthreads();
    }
    const float inv = 1.0f / red[0];
    w[t] = e * inv;
    __syncthreads();

    // ---- pass 2: weighted sum over n; lane streams its d-row with b128 loads
    for (int d = t; d < D_; d += 256) {
        const float* row = kb + (size_t)b * D_ * N_ + (size_t)d * N_;
        float s = 0.0f;
        #pragma unroll 4
        for (int n = 0; n < N_; n += 4) {            // N_ = 196 = 49 * 4
            v4f kv = *(const v4f*)(row + n);
            v4f wv = *(const v4f*)(w + n);
            s += wv.x * kv.x + wv.y * kv.y + wv.z * kv.z + wv.w * kv.w;
        }
        out[b * D_ + d] = s;
    }
}

// ---------------------------------------------------------------------------
extern "C" void kernel_launch(void* const* d_in, const int* in_sizes, int n_in,
                              void* d_out, int out_size, void* d_ws, size_t ws_size,
                              hipStream_t stream)
{
    (void)in_sizes; (void)n_in; (void)out_size; (void)ws_size;

    const float* memory_state   = (const float*)d_in[0];  // [B,D]
    const float* knowledge_base = (const float*)d_in[1];  // [B,D,N]
    const float* ctrl_state     = (const float*)d_in[2];  // [B,D]
    const float* kb_proj        = (const float*)d_in[3];  // [B,D,N]
    const float* W_mem          = (const float*)d_in[4];  // [D,D]
    const float* b_mem          = (const float*)d_in[5];  // [D]
    const float* W_cat          = (const float*)d_in[6];  // [D,2D]
    // d_in[7] = b_cat, d_in[9] = b_attn: constant over n -> softmax invariant.
    const float* W_attn         = (const float*)d_in[8];  // [1,D]

    float* mem_buf = (float*)d_ws;             // B*D   floats (1 MB)
    float* u_buf   = mem_buf + B_ * D_;        // B*2D  floats (2 MB)
    float* out     = (float*)d_out;            // [B,D]

    mem_proj_kernel<<<256, 128, 0, stream>>>(memory_state, W_mem, b_mem, mem_buf);
    u_proj_kernel<<<512, 128, 0, stream>>>(ctrl_state, W_attn, W_cat, u_buf);
    attend_kernel<<<512, 256, 0, stream>>>(knowledge_base, kb_proj,
                                           mem_buf, u_buf, out);
}